// MambaBlock_10273561772162
// MI455X (gfx1250) — compile-verified
//
#include <hip/hip_runtime.h>
#include <hip/hip_bf16.h>

// ---------------------------------------------------------------------------
// Mamba block forward for MI455X (gfx1250, wave32).
// Sizes (fixed by reference): B=2, L=2048, D_MODEL=1024, D_INNER=2048,
// D_STATE=16, D_CONV=4.  M = B*L = 4096.
// Pipeline:
//   1) xz   = x @ W_in^T            (WMMA f32 GEMM, 4096x4096x1024)
//   2) u    = silu(causal dwconv4(xz[:, :2048]) + b)
//   3) xdbl = u @ W_x^T             (skinny N=33: wave-per-output reduction)
//   4) selective scan (sequential in L; thread per (b,d) channel, h[16] regs)
//   5) fused LayerNorm + D*u + silu(z) gating  (in-place on scan output)
//   6) out  = y @ W_out^T           (WMMA f32 GEMM, 4096x1024x2048)
// ---------------------------------------------------------------------------

typedef __attribute__((ext_vector_type(2))) float v2f;
typedef __attribute__((ext_vector_type(8))) float v8f;

#define D_MODEL 1024
#define D_INNER 2048
#define D_STATE 16
#define L_SEQ   2048
#define B_SZ    2
#define M_ROWS  (B_SZ * L_SEQ)   // 4096

// ---------------------------------------------------------------------------
// WMMA f32 GEMM:  C[M,N] = A[M,K] * B[N,K]^T   (A,B row-major)
// Block: 256 threads (8 waves).  Block tile 64(M) x 128(N), K-tile 32.
// Wave tile 32x32 = 2x2 x (16x16) WMMA accumulators.
// Requires M%64==0, N%128==0, K%32==0 (true for all call sites).
// ---------------------------------------------------------------------------
__global__ __launch_bounds__(256)
void gemm_f32_wmma(const float* __restrict__ A, const float* __restrict__ Bm,
                   float* __restrict__ C, int M, int N, int K) {
    __shared__ __align__(16) float As[64][36];   // pad 36: conflict-free frags
    __shared__ __align__(16) float Bs[128][36];

    const int tid  = threadIdx.x;
    const int lane = tid & 31;
    const int w    = tid >> 5;          // wave 0..7
    const int wm   = w >> 2;            // 0..1  (M)
    const int wn   = w & 3;             // 0..3  (N)
    const int lmod = lane & 15;
    const int lhal = lane >> 4;         // 0/1

    const int n0 = blockIdx.x * 128;
    const int m0 = blockIdx.y * 64;

    v8f acc[2][2] = {};

    for (int k0 = 0; k0 < K; k0 += 32) {
        // Stage A tile: 64x32 = 512 float4, 2 per thread
        #pragma unroll
        for (int i = 0; i < 2; ++i) {
            int id = tid + i * 256;
            int r = id >> 3, c4 = id & 7;
            *(float4*)&As[r][c4 * 4] =
                *(const float4*)&A[(size_t)(m0 + r) * K + k0 + c4 * 4];
        }
        // Stage B tile: 128x32 = 1024 float4, 4 per thread
        #pragma unroll
        for (int i = 0; i < 4; ++i) {
            int id = tid + i * 256;
            int r = id >> 3, c4 = id & 7;
            *(float4*)&Bs[r][c4 * 4] =
                *(const float4*)&Bm[(size_t)(n0 + r) * K + k0 + c4 * 4];
        }
        __syncthreads();

        #pragma unroll
        for (int kk = 0; kk < 8; ++kk) {
            const int kc = kk * 4 + lhal * 2;   // A: lanes 0-15 K={0,1}, 16-31 K={2,3}
            v2f a[2], b[2];
            a[0] = *(const v2f*)&As[wm * 32 +      lmod][kc];
            a[1] = *(const v2f*)&As[wm * 32 + 16 + lmod][kc];
            b[0] = *(const v2f*)&Bs[wn * 32 +      lmod][kc];
            b[1] = *(const v2f*)&Bs[wn * 32 + 16 + lmod][kc];
            #pragma unroll
            for (int mi = 0; mi < 2; ++mi)
                #pragma unroll
                for (int ni = 0; ni < 2; ++ni)
                    acc[mi][ni] = __builtin_amdgcn_wmma_f32_16x16x4_f32(
                        false, a[mi], false, b[ni], (short)0, acc[mi][ni],
                        false, false);
        }
        __syncthreads();
    }

    // D layout: VGPR r -> M = r (lanes 0-15) / M = 8+r (lanes 16-31); N = lane%16
    #pragma unroll
    for (int mi = 0; mi < 2; ++mi)
        #pragma unroll
        for (int ni = 0; ni < 2; ++ni)
            #pragma unroll
            for (int r = 0; r < 8; ++r) {
                int gm = m0 + wm * 32 + mi * 16 + lhal * 8 + r;
                int gn = n0 + wn * 32 + ni * 16 + lmod;
                C[(size_t)gm * N + gn] = acc[mi][ni][r];
            }
}

// ---------------------------------------------------------------------------
// u = silu( depthwise causal conv4(xz[..., :D_INNER]) + conv_b )
// One thread per (b,l,d) output element; coalesced over d.
// ---------------------------------------------------------------------------
__global__ __launch_bounds__(256)
void conv_silu_kernel(const float* __restrict__ xz, const float* __restrict__ cw,
                      const float* __restrict__ cb, float* __restrict__ u) {
    size_t id = (size_t)blockIdx.x * 256 + threadIdx.x;  // < B*L*D_INNER
    int d = (int)(id & (D_INNER - 1));
    int l = (int)((id >> 11) & (L_SEQ - 1));
    int b = (int)(id >> 22);
    const float* base = xz + (size_t)b * L_SEQ * (2 * D_INNER) + d;
    float acc = cb[d];
    #pragma unroll
    for (int j = 0; j < 4; ++j) {
        int ll = l - 3 + j;
        if (ll >= 0) acc += base[(size_t)ll * (2 * D_INNER)] * cw[d * 4 + j];
    }
    u[id] = acc / (1.f + __expf(-acc));   // silu
}

// ---------------------------------------------------------------------------
// xdbl[row, e] = sum_d u[row,d] * W_x[e,d]   (row < 4096, e < 33)
// One wave per (row, e); lane-strided dot + shfl reduction.
// ---------------------------------------------------------------------------
__global__ __launch_bounds__(256)
void xdbl_kernel(const float* __restrict__ u, const float* __restrict__ Wx,
                 float* __restrict__ xdbl) {
    int gw   = (int)((blockIdx.x * 256 + threadIdx.x) >> 5);
    int lane = threadIdx.x & 31;
    if (gw >= M_ROWS * 33) return;           // uniform per wave
    int row = gw / 33, e = gw - row * 33;
    const float* ur = u  + (size_t)row * D_INNER;
    const float* wr = Wx + (size_t)e * D_INNER;
    float s = 0.f;
    for (int d = lane; d < D_INNER; d += 32) s += ur[d] * wr[d];
    #pragma unroll
    for (int off = 16; off; off >>= 1) s += __shfl_xor(s, off, 32);
    if (lane == 0) xdbl[(size_t)row * 33 + e] = s;
}

// ---------------------------------------------------------------------------
// Selective scan. Non-linear (per-step clip) => strictly sequential in L.
// Thread = (b, d) channel, h[16] in registers. Per-step uniform x_dbl row
// (dt_raw, B[16], C[16]) staged once into LDS.
// ---------------------------------------------------------------------------
__global__ __launch_bounds__(256)
void scan_kernel(const float* __restrict__ xdbl, const float* __restrict__ u,
                 const float* __restrict__ Wdt, const float* __restrict__ bdt,
                 const float* __restrict__ Alog, float* __restrict__ ybuf) {
    const int tid = threadIdx.x;
    const int d = blockIdx.x * 256 + tid;      // 0..2047
    const int b = blockIdx.y;                  // 0..1

    float A[D_STATE];
    #pragma unroll
    for (int n = 0; n < D_STATE; ++n)
        A[n] = -__expf(fminf(Alog[n], 5.f));

    const float wdt = Wdt[d];
    const float bd  = bdt[d];
    float h[D_STATE];
    #pragma unroll
    for (int n = 0; n < D_STATE; ++n) h[n] = 0.f;

    const float* xd = xdbl + (size_t)b * L_SEQ * 33;
    const float* ub = u    + (size_t)b * L_SEQ * D_INNER + d;
    float*       yb = ybuf + (size_t)b * L_SEQ * D_INNER + d;

    __shared__ float srow[33];

    for (int l = 0; l < L_SEQ; ++l) {
        if (tid < 33) srow[tid] = xd[(size_t)l * 33 + tid];
        __syncthreads();

        float dt = srow[0] * wdt + bd;
        dt = (dt > 20.f) ? dt : log1pf(__expf(dt));       // softplus
        dt = fminf(fmaxf(dt, 1e-4f), 10.f);
        const float ut = ub[(size_t)l * D_INNER];

        float y = 0.f;
        #pragma unroll
        for (int n = 0; n < D_STATE; ++n) {
            float dA  = __expf(fminf(fmaxf(dt * A[n], -20.f), 0.f));
            float dBu = fminf(fmaxf(dt * srow[1 + n] * ut, -10.f), 10.f);
            float hn  = fminf(fmaxf(h[n] * dA + dBu, -100.f), 100.f);
            h[n] = hn;
            y += hn * srow[17 + n];
        }
        yb[(size_t)l * D_INNER] = y;
        __syncthreads();
    }
}

// ---------------------------------------------------------------------------
// Fused LayerNorm(d_inner) + D*u + silu(z) gating, in-place on ybuf.
// One 256-thread block per (b,l) row.
// ---------------------------------------------------------------------------
__global__ __launch_bounds__(256)
void ln_gate_kernel(float* __restrict__ ybuf, const float* __restrict__ u,
                    const float* __restrict__ xz, const float* __restrict__ Dp,
                    const float* __restrict__ g, const float* __restrict__ bta) {
    const int row = blockIdx.x;                // b*L + l
    const int tid = threadIdx.x;
    float* yr = ybuf + (size_t)row * D_INNER;

    float v[8], s = 0.f, s2 = 0.f;
    #pragma unroll
    for (int i = 0; i < 8; ++i) {
        v[i] = yr[tid + i * 256];
        s += v[i]; s2 += v[i] * v[i];
    }
    #pragma unroll
    for (int off = 16; off; off >>= 1) {
        s  += __shfl_xor(s,  off, 32);
        s2 += __shfl_xor(s2, off, 32);
    }
    __shared__ float ws1[8], ws2[8];
    const int lane = tid & 31, w = tid >> 5;
    if (lane == 0) { ws1[w] = s; ws2[w] = s2; }
    __syncthreads();
    if (tid == 0) {
        float t1 = 0.f, t2 = 0.f;
        #pragma unroll
        for (int i = 0; i < 8; ++i) { t1 += ws1[i]; t2 += ws2[i]; }
        ws1[0] = t1; ws2[0] = t2;
    }
    __syncthreads();
    const float mu  = ws1[0] * (1.f / D_INNER);
    float var = ws2[0] * (1.f / D_INNER) - mu * mu;
    var = fmaxf(var, 0.f);
    const float rs = rsqrtf(var + 1e-5f);

    #pragma unroll
    for (int i = 0; i < 8; ++i) {
        const int d = tid + i * 256;
        float yn = (v[i] - mu) * rs * g[d] + bta[d];
        float uu = u[(size_t)row * D_INNER + d];
        float zz = xz[(size_t)row * (2 * D_INNER) + D_INNER + d];
        float sz = zz / (1.f + __expf(-zz));
        yr[d] = (yn + Dp[d] * uu) * sz;
    }
}

// ---------------------------------------------------------------------------
// Host-side launch sequence (all on `stream`, graph-capture safe).
// ---------------------------------------------------------------------------
extern "C" void kernel_launch(void* const* d_in, const int* in_sizes, int n_in,
                              void* d_out, int out_size, void* d_ws, size_t ws_size,
                              hipStream_t stream) {
    const float* x      = (const float*)d_in[0];
    const float* W_in   = (const float*)d_in[1];
    const float* conv_w = (const float*)d_in[2];
    const float* conv_b = (const float*)d_in[3];
    const float* W_x    = (const float*)d_in[4];
    const float* W_dt   = (const float*)d_in[5];
    const float* b_dt   = (const float*)d_in[6];
    const float* A_log  = (const float*)d_in[7];
    const float* D_par  = (const float*)d_in[8];
    const float* W_out  = (const float*)d_in[9];
    const float* ln_g   = (const float*)d_in[10];
    const float* ln_b   = (const float*)d_in[11];
    float* out = (float*)d_out;

    float* ws = (float*)d_ws;
    // workspace layout (floats)
    const size_t XZ_OFF   = 0;                                  // 4096*4096
    const size_t U_OFF    = XZ_OFF + (size_t)M_ROWS * 2 * D_INNER;
    const size_t XDBL_OFF = U_OFF  + (size_t)M_ROWS * D_INNER;  // 4096*33
    const size_t Y_OFF    = XDBL_OFF + (size_t)M_ROWS * 33;
    float* xz   = ws + XZ_OFF;
    float* u    = ws + U_OFF;
    float* xdbl = ws + XDBL_OFF;
    float* ybuf = ws + Y_OFF;

    // 1) xz = x @ W_in^T : M=4096, N=4096, K=1024
    gemm_f32_wmma<<<dim3(2 * D_INNER / 128, M_ROWS / 64), 256, 0, stream>>>(
        x, W_in, xz, M_ROWS, 2 * D_INNER, D_MODEL);

    // 2) u = silu(conv(xz[:, :D_INNER]))
    {
        size_t n = (size_t)M_ROWS * D_INNER;   // 8,388,608
        conv_silu_kernel<<<(unsigned)(n / 256), 256, 0, stream>>>(
            xz, conv_w, conv_b, u);
    }

    // 3) xdbl = u @ W_x^T : one wave per (row, e), 4096*33 waves
    {
        int waves = M_ROWS * 33;
        xdbl_kernel<<<(waves + 7) / 8, 256, 0, stream>>>(u, W_x, xdbl);
    }

    // 4) selective scan: thread per (b, d)
    scan_kernel<<<dim3(D_INNER / 256, B_SZ), 256, 0, stream>>>(
        xdbl, u, W_dt, b_dt, A_log, ybuf);

    // 5) fused LN + D*u + silu(z) gating (in-place on ybuf)
    ln_gate_kernel<<<M_ROWS, 256, 0, stream>>>(ybuf, u, xz, D_par, ln_g, ln_b);

    // 6) out = ybuf @ W_out^T : M=4096, N=1024, K=2048
    gemm_f32_wmma<<<dim3(D_MODEL / 128, M_ROWS / 64), 256, 0, stream>>>(
        ybuf, W_out, out, M_ROWS, D_MODEL, D_INNER);
}